// LinearVectorTransform_18176301597385
// MI455X (gfx1250) — compile-verified
//
#include <hip/hip_runtime.h>

typedef __attribute__((ext_vector_type(2))) float v2f;
typedef __attribute__((ext_vector_type(8))) float v8f;

#define NR_ACTIONS 64
#define VEC        512
#define BATCH      2048
#define WAVES      4
#define NTILE      (WAVES * 16)   // 64 output columns per block
#define APITCH     (VEC + 4)      // LDS pitch pad: stride 516 dwords -> 4-bank skew per row

__global__ __launch_bounds__(128) void lvt_wmma_f32_kernel(
    const float* __restrict__ v,        // [BATCH][VEC]
    const int*   __restrict__ action,   // [BATCH]
    const float* __restrict__ weights,  // [NR_ACTIONS][VEC][VEC]
    float*       __restrict__ out)      // [BATCH][VEC]
{
    __shared__ int   rows[BATCH];          // compacted batch indices for this action
    __shared__ int   count;
    __shared__ float atile[16 * APITCH];   // gathered 16xVEC A tile (padded)

    const int a    = blockIdx.x;           // action handled by this block
    const int n0   = blockIdx.y * NTILE;   // first output column of this block
    const int tid  = threadIdx.x;
    const int lane = tid & 31;
    const int wave = tid >> 5;

    const float* __restrict__ W = weights + (size_t)a * VEC * VEC;

    // ---- prime this block's B slab (VEC rows x NTILE cols = 2 cachelines/row) into L2
    for (int r = tid; r < VEC; r += blockDim.x) {
        __builtin_prefetch(W + (size_t)r * VEC + n0,      0, 1);
        __builtin_prefetch(W + (size_t)r * VEC + n0 + 32, 0, 1);
    }

    // ---- compact the batch rows whose action == a
    if (tid == 0) count = 0;
    __syncthreads();
    for (int b = tid; b < BATCH; b += blockDim.x) {
        if (action[b] == a) {
            int idx = atomicAdd(&count, 1);
            rows[idx] = b;
        }
    }
    __syncthreads();
    const int cnt = count;

    // per-lane WMMA coordinates (fixed across the K loop)
    const int mn    = lane & 15;          // A: M index / B: N index
    const int khalf = (lane >> 4) << 1;   // 0 for lanes 0-15, 2 for lanes 16-31
    const int nw    = n0 + wave * 16;     // this wave's first output column

    for (int m0 = 0; m0 < cnt; m0 += 16) {
        // ---- stage gathered A tile (16 rows x VEC) into LDS, coalesced float4
        for (int i = tid; i < 16 * (VEC / 4); i += blockDim.x) {
            int r = i >> 7;               // i / (VEC/4)
            int c = (i & 127) << 2;       // (i % (VEC/4)) * 4
            int row = (m0 + r < cnt) ? rows[m0 + r] : rows[m0];  // safe pad
            float4 val = *(const float4*)(v + (size_t)row * VEC + c);
            *(float4*)(&atile[r * APITCH + c]) = val;
        }
        __syncthreads();

        // ---- 16x16 f32 accumulator tile, K swept 4 at a time via WMMA
        v8f acc = {0.f, 0.f, 0.f, 0.f, 0.f, 0.f, 0.f, 0.f};
        #pragma unroll 4
        for (int k = 0; k < VEC; k += 4) {
            // A 16x4 layout: lane = M, VGPR0/1 = K = khalf, khalf+1
            v2f afrag = *(const v2f*)(&atile[mn * APITCH + k + khalf]);
            // B 4x16 layout: lane = N, VGPR0/1 = K = khalf, khalf+1
            v2f bfrag;
            bfrag.x = W[(size_t)(k + khalf)     * VEC + nw + mn];
            bfrag.y = W[(size_t)(k + khalf + 1) * VEC + nw + mn];
            acc = __builtin_amdgcn_wmma_f32_16x16x4_f32(
                false, afrag, false, bfrag, (short)0, acc, false, false);
        }

        // ---- store: C/D layout: VGPR i -> M = i (lanes 0-15) / 8+i (lanes 16-31); N = lane&15
        #pragma unroll
        for (int i = 0; i < 8; ++i) {
            int m = (lane < 16) ? i : (8 + i);
            if (m0 + m < cnt) {
                out[(size_t)rows[m0 + m] * VEC + nw + mn] = acc[i];
            }
        }
        __syncthreads();   // protect atile before next M-tile overwrites it
    }
}

extern "C" void kernel_launch(void* const* d_in, const int* in_sizes, int n_in,
                              void* d_out, int out_size, void* d_ws, size_t ws_size,
                              hipStream_t stream) {
    (void)in_sizes; (void)n_in; (void)d_ws; (void)ws_size; (void)out_size;
    const float* v       = (const float*)d_in[0];
    const int*   action  = (const int*)d_in[1];
    const float* weights = (const float*)d_in[2];
    float*       out     = (float*)d_out;

    dim3 grid(NR_ACTIONS, VEC / NTILE);   // 64 x 8 = 512 blocks
    lvt_wmma_f32_kernel<<<grid, 128, 0, stream>>>(v, action, weights, out);
}